// SAUC_for_user_27212912787875
// MI455X (gfx1250) — compile-verified
//
#include <hip/hip_runtime.h>
#include <math.h>

// SAUC loss for MI455X (gfx1250, wave32).
// sigmoid((x_i - y_j)/tau) = 1 / (1 + exp(y_j/tau) * exp(-x_i/tau))
// -> precompute exps once per element; each pair costs 1 v_fma + 1 v_rcp.
// Invalid slots are padded with +inf in LDS so rcp(1 + inf*a) == 0: the
// ragged mask costs zero instructions in the hot loop.
// Pair accumulation runs on the matrix pipe: V_WMMA_F32_16X16X4_F32 with
// A = ones(16x4) sums 64 sigmoid values per instruction into D.

#define TAU       1.0f
#define LOG2E     1.44269504088896340736f
#define PAD_A     224   // inv-exp(pos) array: i reaches p+2, rounded for unroll-4 tail
#define PAD_B     216   // exp(neg) array:     j reaches p+14 (<= 214)
#define UPB       8     // users (waves) per block

typedef __attribute__((ext_vector_type(2))) float v2f;
typedef __attribute__((ext_vector_type(8))) float v8f;

__global__ __launch_bounds__(256) void sauc_user_loss_kernel(
    const float* __restrict__ scores_pos,
    const float* __restrict__ scores_neg,
    const int*   __restrict__ pos_counts,
    float*       __restrict__ user_loss,
    int B, int P) {
  __shared__ float smem[UPB * (PAD_A + PAD_B)];

  const int lane = threadIdx.x & 31;
  const int wave = threadIdx.x >> 5;
  const int user = blockIdx.x * UPB + wave;
  const bool live = (user < B);

  float* a_inv = &smem[wave * (PAD_A + PAD_B)];  // exp(-x_i / tau), inf-padded
  float* b_exp = a_inv + PAD_A;                  // exp(+y_j / tau), inf-padded

  // p is wave-uniform. Select FIRST, readfirstlane LAST so the compiler
  // keeps p in an SGPR: all loop control becomes SALU and EXEC stays
  // all-ones through the WMMAs.
  const int p = __builtin_amdgcn_readfirstlane(live ? pos_counts[user] : 1);

  // Phase 1: per-element transcendentals (2*P per user, coalesced loads),
  // with +inf padding beyond p so ragged masking is free in phase 2.
  if (live) {
    const float* xp = scores_pos + (size_t)user * P;
    const float* yn = scores_neg + (size_t)user * P;
    const float k = LOG2E / TAU;
    for (int i = lane; i < PAD_A; i += 32) {
      float x = (i < P) ? xp[i] : 0.0f;
      a_inv[i] = (i < p) ? __builtin_amdgcn_exp2f(-k * x) : INFINITY;
    }
    for (int i = lane; i < PAD_B; i += 32) {
      float y = (i < P) ? yn[i] : 0.0f;
      b_exp[i] = (i < p) ? __builtin_amdgcn_exp2f(k * y) : INFINITY;
    }
  }
  __syncthreads();  // uniform trip counts; waves only read their own region
  if (!live) return;

  const int half = lane >> 4;   // B-operand K group for this half-wave
  const int jcol = lane & 15;   // this lane's negative-column index

  v8f acc  = {0.f, 0.f, 0.f, 0.f, 0.f, 0.f, 0.f, 0.f};
  v2f ones = {1.f, 1.f};        // A-matrix = ones(16x4)

  // Phase 2: 4(i) x 16(j) tiles; 64 pairs per WMMA.
  // Inner body: 1 ds_load_b64 + 2 v_fma + 2 v_rcp + 1 v_wmma. No masking.
  // Unroll 4 so rcp->wmma TRANS hazards are filled by the next tile's work.
  for (int jb = 0; jb < p; jb += 16) {
    const float bj = b_exp[jb + jcol];   // inf if this column is padding
#pragma unroll 4
    for (int ib = 0; ib < p; ib += 4) {
      const int i0 = ib + 2 * half;      // even -> 8B-aligned ds_load_b64
      v2f ia = *(const v2f*)(a_inv + i0);
      // sigmoid = rcp(1 + b_j * inva_i); padding -> rcp(inf) == 0
      float s0 = __builtin_amdgcn_rcpf(__builtin_fmaf(bj, ia.x, 1.0f));
      float s1 = __builtin_amdgcn_rcpf(__builtin_fmaf(bj, ia.y, 1.0f));
      v2f bmat = {s0, s1};
      // D += ones(16x4) x S(4x16); sum over all D elems = 16 * pair_sum,
      // independent of the exact operand lane layout.
      acc = __builtin_amdgcn_wmma_f32_16x16x4_f32(
          /*neg_a=*/false, ones, /*neg_b=*/false, bmat,
          /*c_mod=*/(short)0, acc, /*reuse_a=*/false, /*reuse_b=*/false);
    }
  }

  // Collapse D: 8 VGPR adds per lane, then 5-step wave32 butterfly.
  float part = acc[0] + acc[1] + acc[2] + acc[3] +
               acc[4] + acc[5] + acc[6] + acc[7];
  for (int off = 16; off > 0; off >>= 1)
    part += __shfl_xor(part, off, 32);

  if (lane == 0) {
    const float pf = (float)p;
    user_loss[user] = 1.0f - (part * (1.0f / 16.0f)) / (pf * pf);
  }
}

// Deterministic fixed-order mean over B per-user losses (no float atomics).
__global__ __launch_bounds__(256) void sauc_mean_kernel(
    const float* __restrict__ user_loss, float* __restrict__ out, int B) {
  __shared__ float sm[256];
  float s = 0.0f;
  for (int i = threadIdx.x; i < B; i += 256) s += user_loss[i];
  sm[threadIdx.x] = s;
  __syncthreads();
  for (int off = 128; off > 0; off >>= 1) {
    if ((int)threadIdx.x < off) sm[threadIdx.x] += sm[threadIdx.x + off];
    __syncthreads();
  }
  if (threadIdx.x == 0) out[0] = sm[0] / (float)B;
}

extern "C" void kernel_launch(void* const* d_in, const int* in_sizes, int n_in,
                              void* d_out, int out_size, void* d_ws, size_t ws_size,
                              hipStream_t stream) {
  const float* scores_pos = (const float*)d_in[0];
  const float* scores_neg = (const float*)d_in[1];
  const int*   pos_counts = (const int*)d_in[2];
  const int B = in_sizes[2];            // 4096
  const int P = in_sizes[0] / B;        // 200
  float* user_loss = (float*)d_ws;      // B floats of scratch

  const int grid = (B + UPB - 1) / UPB;
  sauc_user_loss_kernel<<<grid, 256, 0, stream>>>(
      scores_pos, scores_neg, pos_counts, user_loss, B, P);
  sauc_mean_kernel<<<1, 256, 0, stream>>>(user_loss, (float*)d_out, B);
}